// CrossLayer_3135326126563
// MI455X (gfx1250) — compile-verified
//
#include <hip/hip_runtime.h>
#include <hip/hip_bf16.h>

typedef __attribute__((ext_vector_type(2))) float v2f;
typedef __attribute__((ext_vector_type(8))) float v8f;

#define BN_EPS 1e-5f

// ---------------------------------------------------------------------------
// k_init: mu = 0, rsig = 1 (identity BN for layer 0's lazy normalization)
// ---------------------------------------------------------------------------
__global__ void k_init(float* __restrict__ mu, float* __restrict__ rsig, int D) {
    int d = blockIdx.x * blockDim.x + threadIdx.x;
    if (d < D) { mu[d] = 0.0f; rsig[d] = 1.0f; }
}

// ---------------------------------------------------------------------------
// k_prep: wn[d] = w_l[d] * rsig[d];  c = sum_d mu[d] * wn[d]   (one block)
// ---------------------------------------------------------------------------
__global__ __launch_bounds__(256) void k_prep(const float* __restrict__ w_l,
                                              const float* __restrict__ mu,
                                              const float* __restrict__ rsig,
                                              float* __restrict__ wn,
                                              float* __restrict__ cout, int D) {
    __shared__ float red[256];
    int t = threadIdx.x;
    float partial = 0.0f;
    for (int d = t; d < D; d += 256) {
        float wnv = w_l[d] * rsig[d];
        wn[d] = wnv;
        partial = fmaf(mu[d], wnv, partial);
    }
    red[t] = partial;
    __syncthreads();
    for (int off = 128; off > 0; off >>= 1) {
        if (t < off) red[t] += red[t + off];
        __syncthreads();
    }
    if (t == 0) cout[0] = red[0];
}

// ---------------------------------------------------------------------------
// k_dot_wmma: s[b] = sum_d raw[b,d] * wn[d] - c    (GEMV on the matrix pipe)
//
// V_WMMA_F32_16X16X4_F32, D = A x B + C:
//   A (16x4, f32): all 16 M-rows = wn[k..k+3] (replicated) ->
//       lane<16:  a = {wn[k],   wn[k+1]}   (K=0,1)
//       lane>=16: a = {wn[k+2], wn[k+3]}   (K=2,3)
//   B (4x16, f32): column n = data row n ->
//       lane<16:  b = {raw[row=lane,    k  ], raw[row, k+1]}
//       lane>=16: b = {raw[row=lane-16, k+2], raw[row, k+3]}
//   C row 0 (c[0], lanes 0..15) then holds the 16 row-dots.
// One wave handles 16 rows; 256-thread block = 8 waves = 128 rows.
// Four interleaved accumulators keep the D->C accumulate chain 3 ops apart.
// ---------------------------------------------------------------------------
__global__ __launch_bounds__(256) void k_dot_wmma(const float* __restrict__ raw,
                                                  const float* __restrict__ wn,
                                                  const float* __restrict__ cptr,
                                                  float* __restrict__ s, int D) {
    const int lane    = threadIdx.x & 31;
    const int wave    = threadIdx.x >> 5;
    const int rowBase = blockIdx.x * 128 + wave * 16;
    const int kOff    = (lane >> 4) << 1;          // 0 or 2
    const int row     = rowBase + (lane & 15);
    const float* rp   = raw + (size_t)row * D;

    v8f c0 = {}, c1 = {}, c2 = {}, c3 = {};
    for (int k = 0; k < D; k += 16) {
        v2f a0 = *(const v2f*)(wn + k + kOff);
        v2f b0 = *(const v2f*)(rp + k + kOff);
        v2f a1 = *(const v2f*)(wn + k + 4 + kOff);
        v2f b1 = *(const v2f*)(rp + k + 4 + kOff);
        v2f a2 = *(const v2f*)(wn + k + 8 + kOff);
        v2f b2 = *(const v2f*)(rp + k + 8 + kOff);
        v2f a3 = *(const v2f*)(wn + k + 12 + kOff);
        v2f b3 = *(const v2f*)(rp + k + 12 + kOff);
        c0 = __builtin_amdgcn_wmma_f32_16x16x4_f32(false, a0, false, b0,
                                                   (short)0, c0, false, false);
        c1 = __builtin_amdgcn_wmma_f32_16x16x4_f32(false, a1, false, b1,
                                                   (short)0, c1, false, false);
        c2 = __builtin_amdgcn_wmma_f32_16x16x4_f32(false, a2, false, b2,
                                                   (short)0, c2, false, false);
        c3 = __builtin_amdgcn_wmma_f32_16x16x4_f32(false, a3, false, b3,
                                                   (short)0, c3, false, false);
    }
    float dot = (c0[0] + c1[0]) + (c2[0] + c3[0]);
    if (lane < 16) s[row] = dot - cptr[0];
}

// ---------------------------------------------------------------------------
// k_update: lazy-normalize previous raw, apply cross-layer update, emit new
// raw + per-block partial column sums / sums-of-squares (deterministic).
// 64 rows/block, 256 threads, 8 CONTIGUOUS columns per thread -> all traffic
// is b128 and each block iteration covers exactly one 8 KB row contiguously.
// ---------------------------------------------------------------------------
#define RPB 64

__device__ __forceinline__ float4 upd4(float4 sv, float4 xv, float4 bv,
                                       float4 mv, float4 rv, float sb,
                                       float4& acc, float4& sq) {
    float4 v;
    v.x = fmaf(xv.x, sb, bv.x + (sv.x - mv.x) * rv.x);
    v.y = fmaf(xv.y, sb, bv.y + (sv.y - mv.y) * rv.y);
    v.z = fmaf(xv.z, sb, bv.z + (sv.z - mv.z) * rv.z);
    v.w = fmaf(xv.w, sb, bv.w + (sv.w - mv.w) * rv.w);
    acc.x += v.x; acc.y += v.y; acc.z += v.z; acc.w += v.w;
    sq.x = fmaf(v.x, v.x, sq.x); sq.y = fmaf(v.y, v.y, sq.y);
    sq.z = fmaf(v.z, v.z, sq.z); sq.w = fmaf(v.w, v.w, sq.w);
    return v;
}

__global__ __launch_bounds__(256) void k_update(const float* __restrict__ src,
                                                const float* __restrict__ x0,
                                                const float* __restrict__ bias,
                                                const float* __restrict__ s,
                                                const float* __restrict__ mu,
                                                const float* __restrict__ rsig,
                                                float* __restrict__ dst,
                                                float* __restrict__ psum,
                                                float* __restrict__ psumsq,
                                                int D) {
    const int t = threadIdx.x;
    const int rowBase = blockIdx.x * RPB;
    const int d0 = t * 8;

    const float4 bv0 = *(const float4*)(bias + d0);
    const float4 bv1 = *(const float4*)(bias + d0 + 4);
    const float4 mv0 = *(const float4*)(mu + d0);
    const float4 mv1 = *(const float4*)(mu + d0 + 4);
    const float4 rv0 = *(const float4*)(rsig + d0);
    const float4 rv1 = *(const float4*)(rsig + d0 + 4);

    float4 acc0 = {0.f, 0.f, 0.f, 0.f}, acc1 = {0.f, 0.f, 0.f, 0.f};
    float4 sq0  = {0.f, 0.f, 0.f, 0.f}, sq1  = {0.f, 0.f, 0.f, 0.f};

    for (int r = 0; r < RPB; ++r) {
        const int row = rowBase + r;
        const float sb = s[row];
        const size_t base = (size_t)row * D + d0;
        const float4 sv0 = *(const float4*)(src + base);
        const float4 sv1 = *(const float4*)(src + base + 4);
        const float4 xv0 = *(const float4*)(x0 + base);
        const float4 xv1 = *(const float4*)(x0 + base + 4);
        const float4 v0 = upd4(sv0, xv0, bv0, mv0, rv0, sb, acc0, sq0);
        const float4 v1 = upd4(sv1, xv1, bv1, mv1, rv1, sb, acc1, sq1);
        *(float4*)(dst + base)     = v0;
        *(float4*)(dst + base + 4) = v1;
    }
    const size_t pbase = (size_t)blockIdx.x * D + d0;
    *(float4*)(psum + pbase)       = acc0;
    *(float4*)(psum + pbase + 4)   = acc1;
    *(float4*)(psumsq + pbase)     = sq0;
    *(float4*)(psumsq + pbase + 4) = sq1;
}

// ---------------------------------------------------------------------------
// k_stats: fixed-order reduction over block partials -> mu, rsig (float4)
// ---------------------------------------------------------------------------
__global__ __launch_bounds__(256) void k_stats(const float* __restrict__ psum,
                                               const float* __restrict__ psumsq,
                                               float* __restrict__ mu,
                                               float* __restrict__ rsig,
                                               int D, int nblk, float invB) {
    const int d = (blockIdx.x * blockDim.x + threadIdx.x) * 4;
    float4 sum = {0.f, 0.f, 0.f, 0.f};
    float4 sq  = {0.f, 0.f, 0.f, 0.f};
    for (int i = 0; i < nblk; ++i) {
        const size_t base = (size_t)i * D + d;
        const float4 a = *(const float4*)(psum + base);
        const float4 b = *(const float4*)(psumsq + base);
        sum.x += a.x; sum.y += a.y; sum.z += a.z; sum.w += a.w;
        sq.x  += b.x; sq.y  += b.y; sq.z  += b.z; sq.w  += b.w;
    }
    float4 m, rs;
    m.x = sum.x * invB; m.y = sum.y * invB; m.z = sum.z * invB; m.w = sum.w * invB;
    rs.x = rsqrtf(fmaf(sq.x, invB, -m.x * m.x) + BN_EPS);
    rs.y = rsqrtf(fmaf(sq.y, invB, -m.y * m.y) + BN_EPS);
    rs.z = rsqrtf(fmaf(sq.z, invB, -m.z * m.z) + BN_EPS);
    rs.w = rsqrtf(fmaf(sq.w, invB, -m.w * m.w) + BN_EPS);
    *(float4*)(mu + d)   = m;
    *(float4*)(rsig + d) = rs;
}

// ---------------------------------------------------------------------------
// k_norm: final BN pass, float4 vectorized, in-place safe (elementwise)
// ---------------------------------------------------------------------------
__global__ __launch_bounds__(256) void k_norm(const float* __restrict__ raw,
                                              const float* __restrict__ mu,
                                              const float* __restrict__ rsig,
                                              float* __restrict__ out, int D) {
    const size_t idx = ((size_t)blockIdx.x * blockDim.x + threadIdx.x) * 4;
    const int d = (int)(idx % (size_t)D);
    const float4 v  = *(const float4*)(raw + idx);
    const float4 m  = *(const float4*)(mu + d);
    const float4 rs = *(const float4*)(rsig + d);
    float4 o;
    o.x = (v.x - m.x) * rs.x;
    o.y = (v.y - m.y) * rs.y;
    o.z = (v.z - m.z) * rs.z;
    o.w = (v.w - m.w) * rs.w;
    *(float4*)(out + idx) = o;
}

// ---------------------------------------------------------------------------
extern "C" void kernel_launch(void* const* d_in, const int* in_sizes, int n_in,
                              void* d_out, int out_size, void* d_ws, size_t ws_size,
                              hipStream_t stream) {
    const float* x = (const float*)d_in[0];   // [B, D]
    const float* w = (const float*)d_in[1];   // [L, D]
    const float* b = (const float*)d_in[2];   // [L, D]
    float* out = (float*)d_out;               // [B, D], doubles as "raw" buffer

    const int D = 2048;
    const int B = in_sizes[0] / D;            // 8192
    const int L = in_sizes[1] / D;            // 4
    const int NBLK2 = B / RPB;                // 128 update blocks

    // Workspace carving (floats): s[B] | psum[NBLK2*D] | psumsq[NBLK2*D]
    //                             | mu[D] | rsig[D] | wn[D] | c[1]
    float* ws     = (float*)d_ws;
    float* s_buf  = ws;
    float* psum   = s_buf + B;
    float* psumsq = psum + (size_t)NBLK2 * D;
    float* mu     = psumsq + (size_t)NBLK2 * D;
    float* rsig   = mu + D;
    float* wn     = rsig + D;
    float* cbuf   = wn + D;

    k_init<<<D / 256, 256, 0, stream>>>(mu, rsig, D);

    for (int i = 0; i < L; ++i) {
        const float* src = (i == 0) ? x : out;     // raw matrix from prev layer
        k_prep<<<1, 256, 0, stream>>>(w + (size_t)i * D, mu, rsig, wn, cbuf, D);
        k_dot_wmma<<<B / 128, 256, 0, stream>>>(src, wn, cbuf, s_buf, D);
        k_update<<<NBLK2, 256, 0, stream>>>(src, x, b + (size_t)i * D, s_buf,
                                            mu, rsig, out, psum, psumsq, D);
        k_stats<<<D / (256 * 4), 256, 0, stream>>>(psum, psumsq, mu, rsig, D,
                                                   NBLK2, 1.0f / (float)B);
    }
    // Apply the last layer's BN (was kept lazy) to produce the final output.
    k_norm<<<(int)(((size_t)B * D) / (4 * 256)), 256, 0, stream>>>(out, mu, rsig,
                                                                   out, D);
}